// LSSTransform_Lite_9663676416627
// MI455X (gfx1250) — compile-verified
//
#include <hip/hip_runtime.h>
#include <hip/hip_bf16.h>
#include <math.h>

#define NCAM 6
#define IN_CH 256
#define OUT_CH 80
#define DD 118
#define TOT_CH 198            /* DD + OUT_CH */
#define FH 32
#define FW 88
#define NPIX (FH*FW)          /* 2816 */
#define NXX 360
#define NXY 360
#define OUT_PLANE (NXX*NXY)   /* 129600 */
#define WAVES_PER_BLK 4

typedef __attribute__((ext_vector_type(2))) float v2f;
typedef __attribute__((ext_vector_type(8))) float v8f;

// ---------------------------------------------------------------------------
// 3x3 inverse (adjugate / det), row-major
// ---------------------------------------------------------------------------
__device__ inline void inv3(const float* a, float* o) {
    float c00 = a[4]*a[8] - a[5]*a[7];
    float c01 = a[5]*a[6] - a[3]*a[8];
    float c02 = a[3]*a[7] - a[4]*a[6];
    float det = a[0]*c00 + a[1]*c01 + a[2]*c02;
    float id  = 1.0f / det;
    o[0] = c00*id;
    o[1] = (a[2]*a[7] - a[1]*a[8])*id;
    o[2] = (a[1]*a[5] - a[2]*a[4])*id;
    o[3] = c01*id;
    o[4] = (a[0]*a[8] - a[2]*a[6])*id;
    o[5] = (a[2]*a[3] - a[0]*a[5])*id;
    o[6] = c02*id;
    o[7] = (a[1]*a[6] - a[0]*a[7])*id;
    o[8] = (a[0]*a[4] - a[1]*a[3])*id;
}

// ---------------------------------------------------------------------------
// Per-camera geometry setup: gp[n*24 + ...] =
//   [0..8]  inv(post_rots)   [9..17] rots @ inv(intrins)
//   [18..20] cam2lidar_trans [21..23] post_trans
// ---------------------------------------------------------------------------
__global__ void setup_geom(const float* __restrict__ rots,
                           const float* __restrict__ trans,
                           const float* __restrict__ intr,
                           const float* __restrict__ post_rots,
                           const float* __restrict__ post_trans,
                           float* __restrict__ gp) {
    int n = threadIdx.x;
    if (n >= NCAM) return;
    float invK[9], invPR[9];
    inv3(intr + n*9, invK);
    inv3(post_rots + n*9, invPR);
    const float* R = rots + n*9;
    float* o = gp + n*24;
    for (int i = 0; i < 9; ++i) o[i] = invPR[i];
    for (int i = 0; i < 3; ++i)
        for (int j = 0; j < 3; ++j) {
            float s = 0.f;
            for (int k = 0; k < 3; ++k) s += R[i*3+k] * invK[k*3+j];
            o[9 + i*3 + j] = s;
        }
    for (int i = 0; i < 3; ++i) o[18+i] = trans[n*3+i];
    for (int i = 0; i < 3; ++i) o[21+i] = post_trans[n*3+i];
}

// ---------------------------------------------------------------------------
// GEMM via V_WMMA_F32_16X16X4_F32. Each wave computes one 16x16 tile of
// x[cam, m, p] = sum_k W[m,k] * img[cam,k,p] + bias[m].
// A layout (16x4 f32): lanes0-15 M=lid, v0=K(2*half), v1=K(2*half+1)
// B layout (4x16 f32): lane holds N=lid of rows K=2*half (v0), K=2*half+1 (v1)
// C/D layout: VGPR v, lane l -> C[M = v + 8*(l>>4)][N = l&15]
// Out-of-range M rows: clamp A pointer to last valid row (no predication in
// the hot loop). Store epilogue: block-uniform branch between a branch-free
// path (full tiles, 12/13 of blocks) and a guarded path (last M tile only).
// ---------------------------------------------------------------------------
__global__ __launch_bounds__(WAVES_PER_BLK * 32)
void gemm_wmma(const float* __restrict__ img, const float* __restrict__ W,
               const float* __restrict__ bias, float* __restrict__ x) {
    const int wave  = threadIdx.x >> 5;
    const int lane  = threadIdx.x & 31;
    const int nBase = (blockIdx.x * WAVES_PER_BLK + wave) * 16;  // pixel tile
    const int mBase = blockIdx.y * 16;                           // channel tile
    const int cam   = blockIdx.z;
    const int half  = lane >> 4;
    const int lid   = lane & 15;

    const int mA     = mBase + lid;
    const int mClamp = (mA < TOT_CH) ? mA : (TOT_CH - 1);
    const float* Wrow = W + (size_t)mClamp * IN_CH;
    const float* Bcol = img + (size_t)cam * IN_CH * NPIX + nBase + lid;

    v8f acc = {};
#pragma unroll 8
    for (int kb = 0; kb < IN_CH; kb += 4) {
        const int kA = kb + 2*half;
        v2f a = *(const v2f*)(Wrow + kA);        // global_load_b64, 8B aligned
        v2f b;
        b.x = Bcol[(size_t)kA * NPIX];
        b.y = Bcol[(size_t)(kA + 1) * NPIX];
        acc = __builtin_amdgcn_wmma_f32_16x16x4_f32(
                  false, a, false, b, (short)0, acc, false, false);
    }

    float* xout = x + (size_t)cam * TOT_CH * NPIX + nBase + lid;
    const int mRow = mBase + 8*half;             // first of this lane's 8 rows
    if (mBase + 16 <= TOT_CH) {
        // Full tile: branch-free stores, bias loads batched.
        float bv[8];
#pragma unroll
        for (int v = 0; v < 8; ++v) bv[v] = bias[mRow + v];
#pragma unroll
        for (int v = 0; v < 8; ++v)
            xout[(size_t)(mRow + v) * NPIX] = acc[v] + bv[v];
    } else {
        // Edge tile (last M block only).
#pragma unroll
        for (int v = 0; v < 8; ++v) {
            int m = mRow + v;
            if (m < TOT_CH) xout[(size_t)m * NPIX] = acc[v] + bias[m];
        }
    }
}

// ---------------------------------------------------------------------------
// In-place softmax over the first DD channels of x[cam, :, p]
// ---------------------------------------------------------------------------
__global__ void softmax_depth(float* __restrict__ x) {
    int idx = blockIdx.x * blockDim.x + threadIdx.x;
    if (idx >= NCAM * NPIX) return;
    int cam = idx / NPIX;
    int p   = idx - cam * NPIX;
    float* col = x + (size_t)cam * TOT_CH * NPIX + p;
    float mx = -1e30f;
    for (int o = 0; o < DD; ++o) mx = fmaxf(mx, col[(size_t)o * NPIX]);
    float s = 0.f;
    for (int o = 0; o < DD; ++o) s += __expf(col[(size_t)o * NPIX] - mx);
    float inv = 1.0f / s;
    for (int o = 0; o < DD; ++o) {
        size_t k = (size_t)o * NPIX;
        col[k] = __expf(col[k] - mx) * inv;
    }
}

// ---------------------------------------------------------------------------
// Zero output accumulator
// ---------------------------------------------------------------------------
__global__ void zero_out(float* __restrict__ out, int n) {
    for (int i = blockIdx.x * blockDim.x + threadIdx.x; i < n;
         i += gridDim.x * blockDim.x)
        out[i] = 0.0f;
}

// ---------------------------------------------------------------------------
// Fused frustum geometry + BEV scatter-add.
// Thread = one (cam, d, pixel) point; 80 channel atomics if in-grid.
// p fastest-varying -> feat reads coalesced, lines hot in L2 across d.
// BEV accumulator (41.5 MB) + feat (5.4 MB) stay resident in the 192 MB L2,
// so the ~160M global_atomic_add_f32 resolve at L2 rate.
// ---------------------------------------------------------------------------
__global__ __launch_bounds__(256)
void scatter_bev(const float* __restrict__ x, const float* __restrict__ gp,
                 float* __restrict__ out) {
    int idx = blockIdx.x * blockDim.x + threadIdx.x;
    const int total = NCAM * DD * NPIX;
    if (idx >= total) return;
    int p   = idx % NPIX;
    int t   = idx / NPIX;
    int d   = t % DD;
    int cam = t / DD;
    int w = p % FW, h = p / FW;

    float xf = (float)w * (703.0f / 87.0f);   // linspace(0, IMG_W-1, FW)
    float yf = (float)h * (255.0f / 31.0f);   // linspace(0, IMG_H-1, FH)
    float df = 1.0f + 0.5f * (float)d;        // arange(1.0, 60.0, 0.5)

    const float* g = gp + cam * 24;
    float p0 = xf - g[21], p1 = yf - g[22], p2 = df - g[23];
    float q0 = g[0]*p0 + g[1]*p1 + g[2]*p2;
    float q1 = g[3]*p0 + g[4]*p1 + g[5]*p2;
    float q2 = g[6]*p0 + g[7]*p1 + g[8]*p2;
    float r0 = q0 * q2, r1 = q1 * q2, r2 = q2;
    float s0 = g[ 9]*r0 + g[10]*r1 + g[11]*r2 + g[18];
    float s1 = g[12]*r0 + g[13]*r1 + g[14]*r2 + g[19];
    float s2 = g[15]*r0 + g[16]*r1 + g[17]*r2 + g[20];

    // (pts - (BX - DX/2)) / DX, truncation-cast like .astype(int32)
    int xi = (int)((s0 + 54.0f) * (1.0f / 0.3f));
    int yi = (int)((s1 + 54.0f) * (1.0f / 0.3f));
    int zi = (int)((s2 + 10.0f) * (1.0f / 20.0f));
    if ((unsigned)xi >= (unsigned)NXX) return;
    if ((unsigned)yi >= (unsigned)NXY) return;
    if (zi != 0) return;

    const float* base  = x + (size_t)cam * TOT_CH * NPIX + p;
    float dval = base[(size_t)d * NPIX];
    const float* featp = base + (size_t)DD * NPIX;
    float* ob = out + xi * NXY + yi;
#pragma unroll 4
    for (int c = 0; c < OUT_CH; ++c)
        atomicAdd(ob + (size_t)c * OUT_PLANE, dval * featp[(size_t)c * NPIX]);
}

// ---------------------------------------------------------------------------
extern "C" void kernel_launch(void* const* d_in, const int* in_sizes, int n_in,
                              void* d_out, int out_size, void* d_ws, size_t ws_size,
                              hipStream_t stream) {
    const float* img        = (const float*)d_in[0];
    const float* rots       = (const float*)d_in[1];
    const float* trans      = (const float*)d_in[2];
    const float* intr       = (const float*)d_in[3];
    const float* post_rots  = (const float*)d_in[4];
    const float* post_trans = (const float*)d_in[5];
    const float* Wd         = (const float*)d_in[6];
    const float* bd         = (const float*)d_in[7];
    float* out = (float*)d_out;

    float* gp = (float*)d_ws;        // 6*24 floats
    float* x  = gp + 256;            // 6*198*2816 floats (~13.4 MB)

    setup_geom<<<1, 32, 0, stream>>>(rots, trans, intr, post_rots, post_trans, gp);

    // 176 N-tiles / 4 waves per block = 44 blocks in x; 13 M-tiles; 6 cams
    dim3 ggrid((NPIX / 16) / WAVES_PER_BLK, (TOT_CH + 15) / 16, NCAM);
    gemm_wmma<<<ggrid, WAVES_PER_BLK * 32, 0, stream>>>(img, Wd, bd, x);

    softmax_depth<<<(NCAM * NPIX + 255) / 256, 256, 0, stream>>>(x);

    zero_out<<<4096, 256, 0, stream>>>(out, out_size);

    const int npts = NCAM * DD * NPIX;
    scatter_bev<<<(npts + 255) / 256, 256, 0, stream>>>(x, gp, out);
}